// GNNClassifier_55525337203127
// MI455X (gfx1250) — compile-verified
//
#include <hip/hip_runtime.h>
#include <math.h>

typedef __attribute__((ext_vector_type(2))) float v2f;
typedef __attribute__((ext_vector_type(8))) float v8f;

#define NFEAT 512
#define HIDDEN 16
#define NCLS 2

// ---------------- degree / normalization ----------------

__global__ void k_init_deg(float* __restrict__ deg, int n) {
  int i = blockIdx.x * blockDim.x + threadIdx.x;
  if (i < n) deg[i] = 1.0f;  // self-loop contribution
}

__global__ void k_deg_count(const int* __restrict__ col, float* __restrict__ deg, int E) {
  int e = blockIdx.x * blockDim.x + threadIdx.x;
  if (e < E) atomicAdd(&deg[col[e]], 1.0f);
}

__global__ void k_deg_rsqrt(float* __restrict__ deg, int n) {
  int i = blockIdx.x * blockDim.x + threadIdx.x;
  if (i < n) deg[i] = rsqrtf(deg[i]);  // deg >= 1 always (self-loops)
}

// ---------------- layer-1 GEMM: h1 = x @ W1 (f32 WMMA 16x16x4) ----------------
// One wave32 computes a 16x16 output tile (16 rows of h1, all 16 hidden cols).
// A-layout (32-bit A 16x4): lanes 0-15: VGPR0=K0,VGPR1=K1 ; lanes 16-31: K2,K3
//   -> each lane loads a contiguous float2 from its row of x.
// B-layout mirrors: lane n holds B[k+khalf][n], B[k+khalf+1][n].
// C/D 16x16 f32: VGPR r = row (r + 8*(lane>=16)), col = lane&15.

__global__ __launch_bounds__(256) void k_gemm1(const float* __restrict__ x,
                                               const float* __restrict__ W1,
                                               float* __restrict__ h1,
                                               int n_nodes) {
  const int lane = threadIdx.x & 31;
  const int tile = blockIdx.x * 8 + (threadIdx.x >> 5);
  const int row0 = tile * 16;
  if (row0 >= n_nodes) return;  // wave-uniform exit: EXEC all-ones for WMMA

  const int nsel  = lane & 15;          // M for A-load, N for B/C
  const int khalf = (lane >> 4) << 1;   // 0 or 2

  const float* arow = x + (size_t)(row0 + nsel) * NFEAT + khalf;

  v8f c = {};
  for (int k = 0; k < NFEAT; k += 4) {
    if ((k & 127) == 0) __builtin_prefetch(arow + k + 128, 0, 0);
    v2f a = *(const v2f*)(arow + k);
    v2f b;
    b.x = W1[(k + khalf) * HIDDEN + nsel];
    b.y = W1[(k + khalf + 1) * HIDDEN + nsel];
    // 8 args: (neg_a, A, neg_b, B, c_mod, C, reuse_a, reuse_b)
    c = __builtin_amdgcn_wmma_f32_16x16x4_f32(false, a, false, b, (short)0, c,
                                              false, false);
  }

  const int mbase = row0 + ((lane >> 4) << 3);
#pragma unroll
  for (int r = 0; r < 8; ++r)
    h1[(size_t)(mbase + r) * HIDDEN + nsel] = c[r];
}

// ---------------- aggregation layer 1 ----------------

__global__ void k_agg1_self(const float* __restrict__ h1, const float* __restrict__ dinv,
                            float* __restrict__ agg1, int n) {
  int t = blockIdx.x * blockDim.x + threadIdx.x;  // n*16 threads
  int node = t >> 4;
  if (node < n) {
    float w = dinv[node] * dinv[node];
    agg1[t] = h1[t] * w;
  }
}

__global__ void k_agg1_edges(const int* __restrict__ row, const int* __restrict__ col,
                             const float* __restrict__ h1, const float* __restrict__ dinv,
                             float* __restrict__ agg1, int E) {
  int t = blockIdx.x * blockDim.x + threadIdx.x;  // E*16 threads
  int e = t >> 4;
  int j = t & 15;
  if (e < E) {
    int r = row[e];
    int c = col[e];
    float w = dinv[r] * dinv[c];
    atomicAdd(&agg1[(size_t)c * HIDDEN + j], h1[(size_t)r * HIDDEN + j] * w);
  }
}

// ---------------- ReLU(bias) + layer-2 GEMM (16x2, tiny) ----------------

__global__ void k_relu_gemm2(const float* __restrict__ agg1, const float* __restrict__ b1,
                             const float* __restrict__ W2, float* __restrict__ h2, int n) {
  int node = blockIdx.x * blockDim.x + threadIdx.x;
  if (node >= n) return;
  float acc0 = 0.f, acc1 = 0.f;
#pragma unroll
  for (int j = 0; j < HIDDEN; ++j) {
    float v = fmaxf(agg1[(size_t)node * HIDDEN + j] + b1[j], 0.f);
    acc0 = fmaf(v, W2[j * NCLS + 0], acc0);
    acc1 = fmaf(v, W2[j * NCLS + 1], acc1);
  }
  h2[(size_t)node * NCLS + 0] = acc0;
  h2[(size_t)node * NCLS + 1] = acc1;
}

// ---------------- aggregation layer 2 ----------------

__global__ void k_agg2_self(const float* __restrict__ h2, const float* __restrict__ dinv,
                            float* __restrict__ agg2, int n) {
  int node = blockIdx.x * blockDim.x + threadIdx.x;
  if (node < n) {
    float w = dinv[node] * dinv[node];
    agg2[(size_t)node * NCLS + 0] = h2[(size_t)node * NCLS + 0] * w;
    agg2[(size_t)node * NCLS + 1] = h2[(size_t)node * NCLS + 1] * w;
  }
}

__global__ void k_agg2_edges(const int* __restrict__ row, const int* __restrict__ col,
                             const float* __restrict__ h2, const float* __restrict__ dinv,
                             float* __restrict__ agg2, int E) {
  int e = blockIdx.x * blockDim.x + threadIdx.x;
  if (e < E) {
    int r = row[e];
    int c = col[e];
    float w = dinv[r] * dinv[c];
    atomicAdd(&agg2[(size_t)c * NCLS + 0], h2[(size_t)r * NCLS + 0] * w);
    atomicAdd(&agg2[(size_t)c * NCLS + 1], h2[(size_t)r * NCLS + 1] * w);
  }
}

// ---------------- bias + log_softmax (2 classes) ----------------

__global__ void k_logsoftmax(const float* __restrict__ agg2, const float* __restrict__ b2,
                             float* __restrict__ out, int n) {
  int node = blockIdx.x * blockDim.x + threadIdx.x;
  if (node >= n) return;
  float z0 = agg2[(size_t)node * NCLS + 0] + b2[0];
  float z1 = agg2[(size_t)node * NCLS + 1] + b2[1];
  float m = fmaxf(z0, z1);
  float lse = m + logf(expf(z0 - m) + expf(z1 - m));
  out[(size_t)node * NCLS + 0] = z0 - lse;
  out[(size_t)node * NCLS + 1] = z1 - lse;
}

// ---------------- host-side launch ----------------

extern "C" void kernel_launch(void* const* d_in, const int* in_sizes, int n_in,
                              void* d_out, int out_size, void* d_ws, size_t ws_size,
                              hipStream_t stream) {
  const float* x  = (const float*)d_in[0];
  const int*   ei = (const int*)d_in[1];   // [2, E] int32 (jax x64 disabled)
  const float* W1 = (const float*)d_in[2];
  const float* b1 = (const float*)d_in[3];
  const float* W2 = (const float*)d_in[4];
  const float* b2 = (const float*)d_in[5];

  const int n = in_sizes[0] / NFEAT;  // 100000
  const int E = in_sizes[1] / 2;      // 3200000
  const int* row = ei;
  const int* col = ei + E;

  // workspace carve-up (256B aligned)
  char* ws = (char*)d_ws;
  size_t off = 0;
  auto carve = [&](size_t bytes) -> void* {
    void* p = ws + off;
    off = (off + bytes + 255) & ~(size_t)255;
    return p;
  };
  float* deg  = (float*)carve((size_t)n * sizeof(float));          // becomes dinv in-place
  float* h1   = (float*)carve((size_t)n * HIDDEN * sizeof(float));
  float* agg1 = (float*)carve((size_t)n * HIDDEN * sizeof(float));
  float* h2   = (float*)carve((size_t)n * NCLS * sizeof(float));
  float* agg2 = (float*)carve((size_t)n * NCLS * sizeof(float));
  (void)ws_size;

  const int B = 256;
  auto blocks = [](long long work, int b) { return (unsigned)((work + b - 1) / b); };

  // degree + normalization
  k_init_deg<<<blocks(n, B), B, 0, stream>>>(deg, n);
  k_deg_count<<<blocks(E, B), B, 0, stream>>>(col, deg, E);
  k_deg_rsqrt<<<blocks(n, B), B, 0, stream>>>(deg, n);

  // layer-1 GEMM (independent of degree; wave-per-16-row-tile, 8 waves/block)
  int tiles = (n + 15) / 16;
  k_gemm1<<<blocks(tiles, 8), B, 0, stream>>>(x, W1, h1, n);

  // layer-1 aggregation
  k_agg1_self<<<blocks((long long)n * HIDDEN, B), B, 0, stream>>>(h1, deg, agg1, n);
  k_agg1_edges<<<blocks((long long)E * HIDDEN, B), B, 0, stream>>>(row, col, h1, deg, agg1, E);

  // ReLU + bias + layer-2 GEMM
  k_relu_gemm2<<<blocks(n, B), B, 0, stream>>>(agg1, b1, W2, h2, n);

  // layer-2 aggregation
  k_agg2_self<<<blocks(n, B), B, 0, stream>>>(h2, deg, agg2, n);
  k_agg2_edges<<<blocks(E, B), B, 0, stream>>>(row, col, h2, deg, agg2, E);

  // bias + log_softmax
  k_logsoftmax<<<blocks(n, B), B, 0, stream>>>(agg2, b2, (float*)d_out, n);
}